// TinstanceLayer_74594991997003
// MI455X (gfx1250) — compile-verified
//
#include <hip/hip_runtime.h>
#include <cstdint>
#include <cstddef>

#define NCLS 80
#define NANCH 3
#define MAX_DET 25
#define IOU_THRES 0.7f
#define MAX_WH 7680.0f
#define BS 16

typedef __attribute__((ext_vector_type(2))) float v2f;
typedef __attribute__((ext_vector_type(8))) float v8f;

__device__ __forceinline__ float sigm(float v) {
    return 1.0f / (1.0f + __expf(-v));
}

// ---------------------------------------------------------------------------
// Kernel 1: decode. One wave32 per anchor box. Lane t loads channels t, t+32,
// t+64 (coalesced 128B/wave), wave-reduces max/argmax over the 80 class
// logits (sigmoid monotone => argmax on logits == argmax on sigmoid), then
// lane 0 emits the class-offset xyxy box (float4) and conf = sig(obj)*sig(max).
// ---------------------------------------------------------------------------
__global__ void decode_kernel(const float* __restrict__ x,
                              const float* __restrict__ anc,   // this level's 3x2
                              float* __restrict__ boxes,       // [BS*n] float4
                              float* __restrict__ conf,        // [BS*n]
                              int ny, int nx, int nPerImg) {
    const int lane = threadIdx.x & 31;
    const int wave = blockIdx.x * (blockDim.x >> 5) + (threadIdx.x >> 5);
    const int b = wave / nPerImg;
    const int i = wave - b * nPerImg;
    const int hw = ny * nx;
    const int a = i / hw;
    const int rem = i - a * hw;
    const int gy = rem / nx;
    const int gx = rem - gy * nx;

    const float* p = x + ((size_t)(b * NANCH + a) * hw + rem) * (NCLS + 5);

    float v0 = p[lane];                                     // ch lane   (<32)
    float v1 = p[lane + 32];                                // ch lane+32 (<64)
    float v2 = (lane + 64 < NCLS + 5) ? p[lane + 64] : 0.f; // ch lane+64 (<85)

    // local best class logit; candidates visited in ascending class id so
    // strict '>' reproduces argmax's first-max tie-break.
    float bc = -3.4e38f; int bj = 0x7fffffff;
    if (lane >= 5)              { bc = v0; bj = lane - 5; }
    if (v1 > bc)                { bc = v1; bj = lane + 27; }
    if (lane + 64 < NCLS + 5 && v2 > bc) { bc = v2; bj = lane + 59; }
    for (int m = 16; m >= 1; m >>= 1) {
        float oc = __shfl_xor(bc, m, 32);
        int   oj = __shfl_xor(bj, m, 32);
        if (oc > bc || (oc == bc && oj < bj)) { bc = oc; bj = oj; }
    }
    float l0 = __shfl(v0, 0, 32);
    float l1 = __shfl(v0, 1, 32);
    float l2 = __shfl(v0, 2, 32);
    float l3 = __shfl(v0, 3, 32);
    float ob = __shfl(v0, 4, 32);

    if (lane == 0) {
        float aw = anc[a * 2 + 0];
        float ah = anc[a * 2 + 1];
        float cx = sigm(l0) * 2.0f + ((float)gx - 0.5f);
        float cy = sigm(l1) * 2.0f + ((float)gy - 0.5f);
        float sw = sigm(l2) * 2.0f;
        float sh = sigm(l3) * 2.0f;
        float w = sw * sw * aw;
        float h = sh * sh * ah;
        float off = (float)bj * MAX_WH;
        float4 bb;
        bb.x = cx - 0.5f * w + off;
        bb.y = cy - 0.5f * h + off;
        bb.z = cx + 0.5f * w + off;
        bb.w = cy + 0.5f * h + off;
        ((float4*)boxes)[(size_t)b * nPerImg + i] = bb;
        conf[(size_t)b * nPerImg + i] = sigm(ob) * sigm(bc);
    }
}

// ---------------------------------------------------------------------------
// Kernel 2: greedy NMS, one 1024-thread block per image. conf lives in
// registers, validity is a per-thread bitmask; boxes stream from L2 each
// round. Replaces the reference's argsort + fori_loop with a direct
// (conf, -idx) argmax per round (identical selection incl. tie-breaks).
// ---------------------------------------------------------------------------
#define NMS_THREADS 1024
#define MAXE 19   // ceil(19200 / 1024)

__global__ __launch_bounds__(NMS_THREADS)
void nms_kernel(const float* __restrict__ boxes, const float* __restrict__ conf,
                int* __restrict__ sel, int* __restrict__ selv, int n) {
    const int b = blockIdx.x;
    const int t = threadIdx.x;
    const float4* bx = (const float4*)boxes + (size_t)b * n;
    const float* cf = conf + (size_t)b * n;

    float myc[MAXE];
    unsigned vmask = 0u;
    #pragma unroll
    for (int j = 0; j < MAXE; ++j) {
        int i = t + j * NMS_THREADS;
        if (i < n) { myc[j] = cf[i]; vmask |= (1u << j); }
        else       { myc[j] = -1.0f; }
    }

    __shared__ float  redc[32];
    __shared__ int    redi[32];
    __shared__ float4 s_box;

    const int lane = t & 31;
    const int wv = t >> 5;

    for (int k = 0; k < MAX_DET; ++k) {
        // conf in (0,1) so -1 is a safe "no candidate" sentinel
        float bc = -1.0f; int bi = 0x7fffffff;
        #pragma unroll
        for (int j = 0; j < MAXE; ++j) {
            if (vmask & (1u << j)) {
                float c = myc[j];
                int i = t + j * NMS_THREADS;
                if (c > bc || (c == bc && i < bi)) { bc = c; bi = i; }
            }
        }
        for (int m = 16; m >= 1; m >>= 1) {
            float oc = __shfl_xor(bc, m, 32);
            int   oi = __shfl_xor(bi, m, 32);
            if (oc > bc || (oc == bc && oi < bi)) { bc = oc; bi = oi; }
        }
        if (lane == 0) { redc[wv] = bc; redi[wv] = bi; }
        __syncthreads();
        if (wv == 0) {
            bc = redc[lane]; bi = redi[lane];
            for (int m = 16; m >= 1; m >>= 1) {
                float oc = __shfl_xor(bc, m, 32);
                int   oi = __shfl_xor(bi, m, 32);
                if (oc > bc || (oc == bc && oi < bi)) { bc = oc; bi = oi; }
            }
            if (lane == 0) {
                int has = (bc >= 0.0f) ? 1 : 0;
                int idx = has ? bi : 0;
                sel[b * MAX_DET + k]  = idx;
                selv[b * MAX_DET + k] = has;
                s_box = bx[idx];
            }
        }
        __syncthreads();
        float4 cb = s_box;
        float a1 = (cb.z - cb.x) * (cb.w - cb.y);
        #pragma unroll
        for (int j = 0; j < MAXE; ++j) {
            if (vmask & (1u << j)) {
                int i = t + j * NMS_THREADS;
                float4 o = bx[i];
                float ltx = fmaxf(cb.x, o.x), lty = fmaxf(cb.y, o.y);
                float rbx = fminf(cb.z, o.z), rby = fminf(cb.w, o.w);
                float iw = fmaxf(rbx - ltx, 0.0f), ih = fmaxf(rby - lty, 0.0f);
                float inter = iw * ih;
                float a2 = (o.z - o.x) * (o.w - o.y);
                float iou = inter / (a1 + a2 - inter);
                if (!(iou <= IOU_THRES)) vmask &= ~(1u << j);  // NaN also drops
            }
        }
        __syncthreads();
    }
}

// ---------------------------------------------------------------------------
// Kernel 3: gather + mask via WMMA. out[b,k,c] = selv[k]*feat[b,c,y_k,x_k].
// One wave per 16x16 output tile; the selv mask is applied as
// D = diag(m) x G using 4 chained V_WMMA_F32_16X16X4_F32 (K-slices p=0..3):
//   A_p[i][k] = (i == 4p+k) ? m_i : 0   (16x4, 2 VGPRs)
//   B_p[k][n] = G[4p+k][n]              (4x16, 2 VGPRs)
// Exact in f32: products are x*1 or x*0, one nonzero term per output.
// ---------------------------------------------------------------------------
__global__ void gather_wmma_kernel(const float* __restrict__ feat,
                                   const int* __restrict__ sel,
                                   const int* __restrict__ selv,
                                   float* __restrict__ out,
                                   int ny, int nx, int C) {
    const int lane = threadIdx.x & 31;
    const int wave = blockIdx.x * (blockDim.x >> 5) + (threadIdx.x >> 5);
    const int ctiles = C >> 4;
    const int b  = wave / (2 * ctiles);
    const int r  = (wave / ctiles) & 1;      // row tile: k in [16r, 16r+16)
    const int ct = wave % ctiles;            // col tile: c in [16ct, 16ct+16)

    const int half = lane >> 4;              // 0: lanes 0-15, 1: lanes 16-31
    const int i0   = lane & 15;              // A row (M) / B,D col (N)
    const int c    = ct * 16 + i0;           // channel this lane loads/stores

    int krow = r * 16 + i0;
    float m = (krow < MAX_DET && selv[b * MAX_DET + krow]) ? 1.0f : 0.0f;

    const int kA0 = half * 2;                // K held in VGPR0: 0 or 2
    const int kA1 = half * 2 + 1;            // K held in VGPR1: 1 or 3

    v8f acc = {0.f, 0.f, 0.f, 0.f, 0.f, 0.f, 0.f, 0.f};
    #pragma unroll
    for (int p = 0; p < 4; ++p) {
        v2f A, B;
        A.x = (i0 == 4 * p + kA0) ? m : 0.0f;
        A.y = (i0 == 4 * p + kA1) ? m : 0.0f;
        int k0 = r * 16 + 4 * p + kA0;
        int k1 = r * 16 + 4 * p + kA1;
        int s0 = (k0 < MAX_DET) ? sel[b * MAX_DET + k0] : 0;
        int s1 = (k1 < MAX_DET) ? sel[b * MAX_DET + k1] : 0;
        int gy0 = (s0 / nx) % ny, gx0 = s0 % nx;
        int gy1 = (s1 / nx) % ny, gx1 = s1 % nx;
        B.x = feat[(((size_t)b * C + c) * ny + gy0) * nx + gx0];
        B.y = feat[(((size_t)b * C + c) * ny + gy1) * nx + gx1];
        acc = __builtin_amdgcn_wmma_f32_16x16x4_f32(false, A, false, B,
                                                    (short)0, acc, false, false);
    }
    #pragma unroll
    for (int v = 0; v < 8; ++v) {
        int M = v + 8 * half;
        int k = r * 16 + M;
        if (k < MAX_DET)
            out[((size_t)b * MAX_DET + k) * C + ct * 16 + i0] = acc[v];
    }
}

// ---------------------------------------------------------------------------
extern "C" void kernel_launch(void* const* d_in, const int* in_sizes, int n_in,
                              void* d_out, int out_size, void* d_ws, size_t ws_size,
                              hipStream_t stream) {
    (void)in_sizes; (void)n_in; (void)out_size; (void)ws_size;

    // setup_inputs() dict order: x0, features0, x1, features1, x2, features2, anchors
    const float* x0 = (const float*)d_in[0];
    const float* f0 = (const float*)d_in[1];
    const float* x1 = (const float*)d_in[2];
    const float* f1 = (const float*)d_in[3];
    const float* x2 = (const float*)d_in[4];
    const float* f2 = (const float*)d_in[5];
    const float* anc = (const float*)d_in[6];   // (3,3,2)
    float* out = (float*)d_out;
    char* ws = (char*)d_ws;

    const int n0 = NANCH * 80 * 80;   // 19200
    const int n1 = NANCH * 40 * 40;   //  4800
    const int n2 = NANCH * 20 * 20;   //  1200

    size_t off = 0;
    float* boxes0 = (float*)(ws + off); off += (size_t)BS * n0 * 16;
    float* boxes1 = (float*)(ws + off); off += (size_t)BS * n1 * 16;
    float* boxes2 = (float*)(ws + off); off += (size_t)BS * n2 * 16;
    float* conf0  = (float*)(ws + off); off += (size_t)BS * n0 * 4;
    float* conf1  = (float*)(ws + off); off += (size_t)BS * n1 * 4;
    float* conf2  = (float*)(ws + off); off += (size_t)BS * n2 * 4;
    int* sel0  = (int*)(ws + off); off += (size_t)BS * MAX_DET * 4;
    int* selv0 = (int*)(ws + off); off += (size_t)BS * MAX_DET * 4;
    int* sel1  = (int*)(ws + off); off += (size_t)BS * MAX_DET * 4;
    int* selv1 = (int*)(ws + off); off += (size_t)BS * MAX_DET * 4;
    int* sel2  = (int*)(ws + off); off += (size_t)BS * MAX_DET * 4;
    int* selv2 = (int*)(ws + off); off += (size_t)BS * MAX_DET * 4;

    // decode: one wave per box, 8 waves per 256-thread block
    decode_kernel<<<BS * n0 / 8, 256, 0, stream>>>(x0, anc + 0,  boxes0, conf0, 80, 80, n0);
    decode_kernel<<<BS * n1 / 8, 256, 0, stream>>>(x1, anc + 6,  boxes1, conf1, 40, 40, n1);
    decode_kernel<<<BS * n2 / 8, 256, 0, stream>>>(x2, anc + 12, boxes2, conf2, 20, 20, n2);

    // NMS: one block per image
    nms_kernel<<<BS, NMS_THREADS, 0, stream>>>(boxes0, conf0, sel0, selv0, n0);
    nms_kernel<<<BS, NMS_THREADS, 0, stream>>>(boxes1, conf1, sel1, selv1, n1);
    nms_kernel<<<BS, NMS_THREADS, 0, stream>>>(boxes2, conf2, sel2, selv2, n2);

    // gather + WMMA mask: waves = BS * 2 * (C/16); 4 waves per 128-thread block
    float* out0 = out;
    float* out1 = out0 + (size_t)BS * MAX_DET * 128;
    float* out2 = out1 + (size_t)BS * MAX_DET * 256;
    gather_wmma_kernel<<<BS * 2 * (128 / 16) / 4, 128, 0, stream>>>(f0, sel0, selv0, out0, 80, 80, 128);
    gather_wmma_kernel<<<BS * 2 * (256 / 16) / 4, 128, 0, stream>>>(f1, sel1, selv1, out1, 40, 40, 256);
    gather_wmma_kernel<<<BS * 2 * (512 / 16) / 4, 128, 0, stream>>>(f2, sel2, selv2, out2, 20, 20, 512);
}